// MultiHeadAttentionLayer_64037962384023
// MI455X (gfx1250) — compile-verified
//
#include <hip/hip_runtime.h>
#include <math.h>

// Problem sizes (fixed by the reference)
#define NN 50000
#define EE 800000
#define FF 128
#define HH 8
#define DD 16

#define LDSPAD 132   // 16x128 tile padded to avoid LDS bank conflicts (132 % 64 = 4)

typedef __attribute__((ext_vector_type(2))) float v2f;
typedef __attribute__((ext_vector_type(8))) float v8f;

// ---- CDNA5 async Global->LDS copy (16B per lane), tracked by ASYNCcnt ----
// GLOBAL_LOAD_ASYNC_TO_LDS_B128, GV mode: vdst = LDS address VGPR, vaddr = 64b
// global address VGPR pair, no saddr ("off"). Low 32 bits of a generic shared
// pointer are the LDS offset (ISA 10.2 aperture rules).
__device__ __forceinline__ void async_copy_b128(float* lds_dst, const float* gsrc) {
  unsigned loff = (unsigned)(uintptr_t)lds_dst;
  asm volatile("global_load_async_to_lds_b128 %0, %1, off"
               :: "v"(loff), "v"(gsrc)
               : "memory");
}
__device__ __forceinline__ void wait_async0() {
  asm volatile("s_wait_asynccnt 0" ::: "memory");
}

// Compute one 16x16 fp32 output tile:  D = A(16x128, LDS) * B(128x128 row-major, cols n0..n0+15)
// using V_WMMA_F32_16X16X4_F32 over a K=128 loop (32 WMMAs).
// A-fragment layout (ISA 7.12.2, 32-bit A 16x4): lane%16 = M row; half (lane>=16)
// selects K pair {k+2,k+3} vs {k,k+1}. B assumed symmetric (K row striped per half).
__device__ __forceinline__ v8f wmma_tile_f32(const float* As,
                                             const float* __restrict__ Bmat,
                                             int n0, int lane) {
  v8f acc = {};
  const int half = lane >> 4;
  const int l16  = lane & 15;
  const int arow = l16 * LDSPAD;
#pragma unroll
  for (int k = 0; k < FF; k += 4) {
    const int kk = k + 2 * half;
    v2f a, b;
    a.x = As[arow + kk];
    a.y = As[arow + kk + 1];
    b.x = Bmat[(kk    ) * FF + n0 + l16];
    b.y = Bmat[(kk + 1) * FF + n0 + l16];
    acc = __builtin_amdgcn_wmma_f32_16x16x4_f32(false, a, false, b,
                                                (short)0, acc, false, false);
  }
  return acc;
}

// ---------------------------------------------------------------- zero
__global__ void mha_zero_kernel(float* __restrict__ hout, float* __restrict__ z) {
  const int idx = blockIdx.x * blockDim.x + threadIdx.x;
  if (idx < NN * FF) hout[idx] = 0.0f;
  if (idx < NN * HH) z[idx]    = 0.0f;
}

// ---------------------------------------------------------------- Q/K/V projection
// Block: 256 threads (8 waves) handles 16 nodes. Wave w owns columns [16w,16w+16).
__global__ void mha_qkv_kernel(const float* __restrict__ node_feats,
                               const float* __restrict__ Wq,
                               const float* __restrict__ Wk,
                               const float* __restrict__ Wv,
                               float* __restrict__ Qb,
                               float* __restrict__ Kb,
                               float* __restrict__ Vb) {
  __shared__ __align__(16) float As[16 * LDSPAD];
  const int row0 = blockIdx.x * 16;
  const int tid  = threadIdx.x;

  // Stage 16x128 A tile into LDS via async DMA (2x b128 per lane)
  {
    const int r = tid >> 4;
    const int c = (tid & 15) * 8;
    const float* sp = node_feats + (size_t)(row0 + r) * FF + c;
    float* dp = &As[r * LDSPAD + c];
    async_copy_b128(dp,     sp);
    async_copy_b128(dp + 4, sp + 4);
    wait_async0();
  }
  __syncthreads();

  const int wave = tid >> 5;
  const int lane = tid & 31;
  const int n0   = wave * 16;
  const int l16  = lane & 15;
  const int half = lane >> 4;

  // Q
  {
    v8f acc = wmma_tile_f32(As, Wq, n0, lane);
#pragma unroll
    for (int v = 0; v < 8; ++v)
      Qb[(row0 + v + 8 * half) * FF + n0 + l16] = acc[v];
  }
  // K
  {
    v8f acc = wmma_tile_f32(As, Wk, n0, lane);
#pragma unroll
    for (int v = 0; v < 8; ++v)
      Kb[(row0 + v + 8 * half) * FF + n0 + l16] = acc[v];
  }
  // V
  {
    v8f acc = wmma_tile_f32(As, Wv, n0, lane);
#pragma unroll
    for (int v = 0; v < 8; ++v)
      Vb[(row0 + v + 8 * half) * FF + n0 + l16] = acc[v];
  }
}

// ---------------------------------------------------------------- edge kernel
// Block: 256 threads handles 16 edges; wave w == head w (16 cols == one head,
// so the D-dim softmax sum is a 16-lane shuffle butterfly inside the wave).
__global__ void mha_edge_kernel(const float* __restrict__ edge_feats,
                                const float* __restrict__ We,
                                const float* __restrict__ Qb,
                                const float* __restrict__ Kb,
                                const float* __restrict__ Vb,
                                const int* __restrict__ src,
                                const int* __restrict__ dst,
                                float* __restrict__ e_out,
                                float* __restrict__ wV,
                                float* __restrict__ z) {
  __shared__ __align__(16) float As[16 * LDSPAD];
  __shared__ int s_src[16];
  __shared__ int s_dst[16];
  const int e0  = blockIdx.x * 16;
  const int tid = threadIdx.x;

  // Async-stage the 16x128 edge_feats tile into LDS
  {
    const int r = tid >> 4;
    const int c = (tid & 15) * 8;
    const float* sp = edge_feats + (size_t)(e0 + r) * FF + c;
    float* dp = &As[r * LDSPAD + c];
    async_copy_b128(dp,     sp);
    async_copy_b128(dp + 4, sp + 4);
  }
  if (tid < 16) {
    const int sN = src[e0 + tid];
    const int dN = dst[e0 + tid];
    s_src[tid] = sN;
    s_dst[tid] = dN;
    // warm the L2-resident gather targets (global_prefetch_b8)
    __builtin_prefetch(&Kb[sN * FF], 0, 3);
    __builtin_prefetch(&Vb[sN * FF], 0, 3);
    __builtin_prefetch(&Qb[dN * FF], 0, 3);
  }
  wait_async0();
  __syncthreads();

  const int wave = tid >> 5;        // head index
  const int lane = tid & 31;
  const int n0   = wave * 16;
  const int l16  = lane & 15;
  const int half = lane >> 4;
  const int nglob = n0 + l16;       // feature column (head*16 + d)

  v8f pe = wmma_tile_f32(As, We, n0, lane);   // proj_e tile

  const float inv_sqrt_d = 0.25f;   // 1/sqrt(16)
#pragma unroll
  for (int v = 0; v < 8; ++v) {
    const int m  = v + 8 * half;    // edge within tile (per half-wave)
    const int sN = s_src[m];
    const int dN = s_dst[m];

    float kq = Kb[sN * FF + nglob] * Qb[dN * FF + nglob] * inv_sqrt_d;
    kq = fminf(fmaxf(kq, -5.0f), 5.0f);
    const float score = kq * pe[v];
    e_out[(e0 + m) * FF + nglob] = score;

    // sum over D=16 within this head: butterfly across the 16-lane half
    float ssum = score;
    ssum += __shfl_xor(ssum, 1, 32);
    ssum += __shfl_xor(ssum, 2, 32);
    ssum += __shfl_xor(ssum, 4, 32);
    ssum += __shfl_xor(ssum, 8, 32);
    ssum = fminf(fmaxf(ssum, -5.0f), 5.0f);
    const float s = __expf(ssum);

    const float val = Vb[sN * FF + nglob] * s;
    atomicAdd(&wV[dN * FF + nglob], val);            // L2-resident scatter
    if (l16 == 0) atomicAdd(&z[dN * HH + wave], s);  // once per (edge, head)
  }
}

// ---------------------------------------------------------------- normalize
__global__ void mha_norm_kernel(float* __restrict__ hout, const float* __restrict__ z) {
  const int idx = blockIdx.x * blockDim.x + threadIdx.x;
  if (idx >= NN * FF) return;
  const int node = idx / FF;
  const int head = (idx % FF) / DD;
  hout[idx] = hout[idx] / (z[node * HH + head] + 1e-6f);
}

// ---------------------------------------------------------------- launch
extern "C" void kernel_launch(void* const* d_in, const int* in_sizes, int n_in,
                              void* d_out, int out_size, void* d_ws, size_t ws_size,
                              hipStream_t stream) {
  const float* node_feats = (const float*)d_in[0];
  const float* edge_feats = (const float*)d_in[1];
  const float* Wq         = (const float*)d_in[2];
  const float* Wk         = (const float*)d_in[3];
  const float* Wv         = (const float*)d_in[4];
  const float* We         = (const float*)d_in[5];
  const int*   src        = (const int*)d_in[6];
  const int*   dst        = (const int*)d_in[7];

  float* hout = (float*)d_out;            // [N, H*D] accumulator then result
  float* eout = hout + (size_t)NN * FF;   // [E, H*D]

  float* Qb = (float*)d_ws;               // [N,128]
  float* Kb = Qb + (size_t)NN * FF;       // [N,128]
  float* Vb = Kb + (size_t)NN * FF;       // [N,128]
  float* zb = Vb + (size_t)NN * FF;       // [N,8]

  mha_zero_kernel<<<(NN * FF + 255) / 256, 256, 0, stream>>>(hout, zb);
  mha_qkv_kernel<<<NN / 16, 256, 0, stream>>>(node_feats, Wq, Wk, Wv, Qb, Kb, Vb);
  mha_edge_kernel<<<EE / 16, 256, 0, stream>>>(edge_feats, We, Qb, Kb, Vb,
                                               src, dst, eout, hout, zb);
  mha_norm_kernel<<<(NN * FF + 255) / 256, 256, 0, stream>>>(hout, zb);
}